// GAT_34282428956965
// MI455X (gfx1250) — compile-verified
//
#include <hip/hip_runtime.h>
#include <hip/hip_bf16.h>
#include <math.h>

typedef float v2f __attribute__((ext_vector_type(2)));
typedef float v8f __attribute__((ext_vector_type(8)));

#define N_NODES   1000
#define M_EDGES   8
#define E_EDGES   (N_NODES * M_EDGES)   // 8000
#define T_ROWS    (N_NODES + E_EDGES)   // 9000
#define HDIM      64
#define NEG_SLOPE 0.2f

__device__ __forceinline__ float leaky(float v) { return v > 0.f ? v : NEG_SLOPE * v; }

// ---------------------------------------------------------------------------
// WMMA f32 GEMM:  C(rows x 64) = A(rows x K) @ B(K x 64)  (+ optional bias)
// One wave per 16-row strip, 4 col tiles => 4x v_wmma_f32_16x16x4_f32 per k-step.
// A frag (16x4 f32): lane L holds A[L&15, kb] , A[L&15, kb+1], kb = (L>>4)*2.
// B frag (4x16 f32): lane L, vgpr v holds B[kb+v, L&15].
// C frag (16x16 f32): lane L, vgpr v holds C[v + 8*(L>>4), L&15].
// ---------------------------------------------------------------------------
#define GEMM_WAVES 4
__global__ void gemm64_wmma(const float* __restrict__ A, const float* __restrict__ B,
                            const float* __restrict__ bias, float* __restrict__ C,
                            int rows, int K) {
  const int lane = threadIdx.x & 31;
  const int wave = threadIdx.x >> 5;
  const int tile = blockIdx.x * GEMM_WAVES + wave;
  const int row0 = tile * 16;
  if (row0 >= rows) return;                 // wave-uniform: EXEC stays all-ones

  const int m     = lane & 15;
  const int khalf = (lane >> 4) << 1;       // 0 or 2
  int arow = row0 + m;
  if (arow >= rows) arow = rows - 1;        // clamp loads; padded rows never stored
  const float* __restrict__ Arow = A + (size_t)arow * K;

  v8f c0 = {}, c1 = {}, c2 = {}, c3 = {};
  for (int k = 0; k < K; k += 4) {
    v2f a;
    a.x = Arow[k + khalf + 0];
    a.y = Arow[k + khalf + 1];
    const float* __restrict__ Bp = B + (size_t)(k + khalf) * HDIM + m;
    v2f b0, b1, b2, b3;
    b0.x = Bp[ 0]; b0.y = Bp[HDIM +  0];
    b1.x = Bp[16]; b1.y = Bp[HDIM + 16];
    b2.x = Bp[32]; b2.y = Bp[HDIM + 32];
    b3.x = Bp[48]; b3.y = Bp[HDIM + 48];
    c0 = __builtin_amdgcn_wmma_f32_16x16x4_f32(false, a, false, b0, (short)0, c0, false, false);
    c1 = __builtin_amdgcn_wmma_f32_16x16x4_f32(false, a, false, b1, (short)0, c1, false, false);
    c2 = __builtin_amdgcn_wmma_f32_16x16x4_f32(false, a, false, b2, (short)0, c2, false, false);
    c3 = __builtin_amdgcn_wmma_f32_16x16x4_f32(false, a, false, b3, (short)0, c3, false, false);
  }

  const int rowoff = (lane >> 4) << 3;      // 0 or 8
  const float bb0 = bias ? bias[m +  0] : 0.f;
  const float bb1 = bias ? bias[m + 16] : 0.f;
  const float bb2 = bias ? bias[m + 32] : 0.f;
  const float bb3 = bias ? bias[m + 48] : 0.f;
  #pragma unroll
  for (int v = 0; v < 8; ++v) {
    int r = row0 + rowoff + v;
    if (r < rows) {
      float* __restrict__ Crow = C + (size_t)r * HDIM + m;
      Crow[ 0] = c0[v] + bb0;
      Crow[16] = c1[v] + bb1;
      Crow[32] = c2[v] + bb2;
      Crow[48] = c3[v] + bb3;
    }
  }
}

// ---------------------------------------------------------------------------
// Per-row dots: s[r] = Xs[r,:]·a_src ; t[r] = Xs[r,:]·a_dst  (wave per row)
// ---------------------------------------------------------------------------
#define DOT_WAVES 8
__global__ void row_dots_kernel(const float* __restrict__ Xs, const float* __restrict__ asrc,
                                const float* __restrict__ adst, float* __restrict__ s_arr,
                                float* __restrict__ t_arr) {
  const int lane = threadIdx.x & 31;
  const int wave = threadIdx.x >> 5;
  const int row = blockIdx.x * DOT_WAVES + wave;
  if (row >= T_ROWS) return;
  const float* __restrict__ xr = Xs + (size_t)row * HDIM;
  const float x0 = xr[lane], x1 = xr[lane + 32];
  float ps = x0 * asrc[lane] + x1 * asrc[lane + 32];
  float pt = x0 * adst[lane] + x1 * adst[lane + 32];
  #pragma unroll
  for (int off = 16; off > 0; off >>= 1) {
    ps += __shfl_xor(ps, off, 32);
    pt += __shfl_xor(pt, off, 32);
  }
  if (lane == 0) { s_arr[row] = ps; t_arr[row] = pt; }
}

// ---------------------------------------------------------------------------
// Node-row diagonal softmax weight. Wave per node: online softmax across
// {self, 8 out-edges} + lane-strided scan of in-edges (deterministic order).
// ---------------------------------------------------------------------------
#define ANODE_WAVES 8
__global__ void alpha_node_kernel(const float* __restrict__ s_arr, const float* __restrict__ t_arr,
                                  const int* __restrict__ eidx, float* __restrict__ alpha) {
  const int lane = threadIdx.x & 31;
  const int wave = threadIdx.x >> 5;
  const int n = blockIdx.x * ANODE_WAVES + wave;
  if (n >= N_NODES) return;
  const float sn = s_arr[n];
  float mx = -3.0e38f, ssum = 0.f;
  if (lane < 9) {                                  // self + own 8 out-edge rows
    const int k = (lane == 0) ? n : (N_NODES + n * M_EDGES + (lane - 1));
    mx = leaky(sn + t_arr[k]);
    ssum = 1.f;
  }
  for (int e = lane; e < E_EDGES; e += 32) {       // in-edges (skip own out-edges)
    if (eidx[e] == n && (e >> 3) != n) {
      const float z = leaky(sn + t_arr[N_NODES + e]);
      if (z > mx) { ssum = ssum * __expf(mx - z) + 1.f; mx = z; }
      else        { ssum += __expf(z - mx); }
    }
  }
  #pragma unroll
  for (int off = 16; off > 0; off >>= 1) {         // combine (m, s) pairs
    const float mo = __shfl_xor(mx, off, 32);
    const float so = __shfl_xor(ssum, off, 32);
    const float mn = fmaxf(mx, mo);
    ssum = ssum * __expf(mx - mn) + so * __expf(mo - mn);
    mx = mn;
  }
  if (lane == 0) {
    const float zs = leaky(sn + t_arr[n]);
    alpha[n] = __expf(zs - mx) / ssum;
  }
}

// ---------------------------------------------------------------------------
// Edge-row diagonal softmax weight: neighbors = {self, src, dst} (dedup src==dst)
// ---------------------------------------------------------------------------
__global__ void alpha_edge_kernel(const float* __restrict__ s_arr, const float* __restrict__ t_arr,
                                  const int* __restrict__ eidx, float* __restrict__ alpha) {
  const int e = blockIdx.x * blockDim.x + threadIdx.x;
  if (e >= E_EDGES) return;
  const int row = N_NODES + e;
  const int i = e >> 3;           // src node
  const int j = eidx[e];          // dst node
  const float sr = s_arr[row];
  const float zs = leaky(sr + t_arr[row]);
  const float zi = leaky(sr + t_arr[i]);
  float mx = fmaxf(zs, zi);
  float ssum;
  if (j != i) {
    const float zj = leaky(sr + t_arr[j]);
    mx = fmaxf(mx, zj);
    ssum = __expf(zs - mx) + __expf(zi - mx) + __expf(zj - mx);
  } else {
    ssum = __expf(zs - mx) + __expf(zi - mx);
  }
  alpha[row] = __expf(zs - mx) / ssum;
}

// ---------------------------------------------------------------------------
// out[r,c] = act(alpha[r] * Xs[r,c])
// ---------------------------------------------------------------------------
__global__ void scale_kernel(const float* __restrict__ Xs, const float* __restrict__ alpha,
                             float* __restrict__ out, int rows, int relu) {
  const int id = blockIdx.x * blockDim.x + threadIdx.x;
  if (id >= rows * HDIM) return;
  const int r = id >> 6;
  float v = alpha[r] * Xs[id];
  if (relu) v = fmaxf(v, 0.f);
  out[id] = v;
}

// ---------------------------------------------------------------------------
extern "C" void kernel_launch(void* const* d_in, const int* in_sizes, int n_in,
                              void* d_out, int out_size, void* d_ws, size_t ws_size,
                              hipStream_t stream) {
  const float* node_in  = (const float*)d_in[0];
  const float* edge_fea = (const float*)d_in[1];
  const int*   eidx     = (const int*)  d_in[2];
  const float* Wn = (const float*)d_in[3];
  const float* bn = (const float*)d_in[4];
  const float* We = (const float*)d_in[5];
  const float* be = (const float*)d_in[6];
  const float* W[3]    = {(const float*)d_in[7],  (const float*)d_in[10], (const float*)d_in[13]};
  const float* asrc[3] = {(const float*)d_in[8],  (const float*)d_in[11], (const float*)d_in[14]};
  const float* adst[3] = {(const float*)d_in[9],  (const float*)d_in[12], (const float*)d_in[15]};
  float* out = (float*)d_out;

  // workspace layout (floats): x[T*64] | xs[T*64] | s[T] | t[T] | alpha[T]  (~4.7 MB)
  float* ws    = (float*)d_ws;
  const size_t T64 = (size_t)T_ROWS * HDIM;
  float* x     = ws;
  float* xs    = x + T64;
  float* s_arr = xs + T64;
  float* t_arr = s_arr + T_ROWS;
  float* alpha = t_arr + T_ROWS;

  const dim3 gblk(32 * GEMM_WAVES);

  // embeddings: x[0:N] = node_in @ Wn + bn ; x[N:T] = edge_fea @ We + be
  {
    const int tilesN = (N_NODES + 15) / 16;   // 63
    gemm64_wmma<<<(tilesN + GEMM_WAVES - 1) / GEMM_WAVES, gblk, 0, stream>>>(
        node_in, Wn, bn, x, N_NODES, 32);
    const int tilesE = (E_EDGES + 15) / 16;   // 500
    gemm64_wmma<<<(tilesE + GEMM_WAVES - 1) / GEMM_WAVES, gblk, 0, stream>>>(
        edge_fea, We, be, x + (size_t)N_NODES * HDIM, E_EDGES, 32);
  }

  const int tilesT = (T_ROWS + 15) / 16;      // 563
  for (int l = 0; l < 3; ++l) {
    gemm64_wmma<<<(tilesT + GEMM_WAVES - 1) / GEMM_WAVES, gblk, 0, stream>>>(
        x, W[l], nullptr, xs, T_ROWS, HDIM);
    row_dots_kernel<<<(T_ROWS + DOT_WAVES - 1) / DOT_WAVES, dim3(32 * DOT_WAVES), 0, stream>>>(
        xs, asrc[l], adst[l], s_arr, t_arr);
    alpha_node_kernel<<<(N_NODES + ANODE_WAVES - 1) / ANODE_WAVES, dim3(32 * ANODE_WAVES), 0, stream>>>(
        s_arr, t_arr, eidx, alpha);
    if (l < 2) {
      alpha_edge_kernel<<<(E_EDGES + 255) / 256, dim3(256), 0, stream>>>(s_arr, t_arr, eidx, alpha);
      scale_kernel<<<((int)T64 + 255) / 256, dim3(256), 0, stream>>>(xs, alpha, x, T_ROWS, 1);
    } else {
      scale_kernel<<<(N_NODES * HDIM + 255) / 256, dim3(256), 0, stream>>>(xs, alpha, out, N_NODES, 0);
    }
  }
}